// AdaptiveFeaturePooling_44367012168181
// MI455X (gfx1250) — compile-verified
//
#include <hip/hip_runtime.h>
#include <cstdint>

#define NUM_GRAPHS 1024
#define D 128
#define TILE 32                    // rows per LDS buffer (16 KB)
#define BLOCK 256                  // threads per block = 8 wave32s
#define WAVES 8
#define ROWS_PER_BLOCK 256
#define TILE_BYTES (TILE * D * 4)                       // 16384
#define CHUNKS_PER_THREAD (TILE_BYTES / (BLOCK * 16))   // 4 x b128 per thread per tile

typedef int v4i_t __attribute__((ext_vector_type(4)));
typedef __attribute__((address_space(1))) v4i_t gv4i_t;  // global int4
typedef __attribute__((address_space(3))) v4i_t lv4i_t;  // LDS int4

#if __has_builtin(__builtin_amdgcn_global_load_async_to_lds_b128)
#define HAVE_ASYNC 1
#else
#define HAVE_ASYNC 0
#endif

__device__ __forceinline__ void async_copy_b128(const void* g, void* l) {
#if HAVE_ASYNC
  __builtin_amdgcn_global_load_async_to_lds_b128((gv4i_t*)(void*)g, (lv4i_t*)l, 0, 0);
#else
  *(float4*)l = *(const float4*)g;   // fallback: plain load/store through LDS
#endif
}

template <int CNT>
__device__ __forceinline__ void wait_async() {
#if HAVE_ASYNC
#if __has_builtin(__builtin_amdgcn_s_wait_asynccnt)
  __builtin_amdgcn_s_wait_asynccnt(CNT);
#else
  asm volatile("s_wait_asynccnt %0" ::"n"(CNT));
#endif
#endif
}

// Order-independent float atomic max via sign-split integer atomics.
// ws_max is initialized to -inf (0xFF800000): int view = very negative (signed
// max works for v>=0); uint view = very large (unsigned min works for v<0).
__device__ __forceinline__ void atomicMaxF(float* addr, float v) {
  if (v >= 0.0f)
    atomicMax((int*)addr, __float_as_int(v));
  else
    atomicMin((unsigned int*)addr, __float_as_uint(v));
}

__global__ __launch_bounds__(BLOCK) void afp_init_kernel(float* ws, int total) {
  int idx = blockIdx.x * BLOCK + threadIdx.x;
  if (idx >= total) return;
  // layout: [0, G*D) sums = 0 ; [G*D, 2*G*D) maxes = -inf ; [2*G*D, +G) counts = 0
  float v = 0.0f;
  if (idx >= NUM_GRAPHS * D && idx < 2 * NUM_GRAPHS * D) v = -__builtin_inff();
  ws[idx] = v;
}

__global__ __launch_bounds__(BLOCK) void afp_pool_kernel(
    const float* __restrict__ x, const int* __restrict__ batch,
    const float* __restrict__ attn_w, const float* __restrict__ attn_b,
    float* __restrict__ ws_sum, float* __restrict__ ws_max,
    float* __restrict__ ws_cnt, int N) {
  __shared__ float buf[2][TILE * D];

  const int tid = threadIdx.x;
  const int lane = tid & 31;
  const int wave = tid >> 5;

  const long long row0 = (long long)blockIdx.x * ROWS_PER_BLOCK;
  long long rowEndLL = row0 + ROWS_PER_BLOCK;
  if (rowEndLL > N) rowEndLL = N;
  const int rowsHere = (int)(rowEndLL - row0);
  if (rowsHere <= 0) return;
  const int ntiles = (rowsHere + TILE - 1) / TILE;

  // gate weights: lane owns 4 contiguous columns
  const float4 w4 = *(const float4*)(attn_w + lane * 4);
  const float bias = attn_b[0];

  // per-(wave,lane) running-segment accumulators
  float sx = 0.f, sy = 0.f, sz = 0.f, sw = 0.f;
  float mx = -__builtin_inff(), my = -__builtin_inff(),
        mz = -__builtin_inff(), mw = -__builtin_inff();
  float cnt = 0.f;
  int curSeg = -1;

  auto flush = [&]() {
    if (curSeg >= 0 && cnt > 0.f) {
      float* sp = ws_sum + (size_t)curSeg * D + lane * 4;
      atomicAdd(sp + 0, sx);
      atomicAdd(sp + 1, sy);
      atomicAdd(sp + 2, sz);
      atomicAdd(sp + 3, sw);
      float* mp = ws_max + (size_t)curSeg * D + lane * 4;
      atomicMaxF(mp + 0, mx);
      atomicMaxF(mp + 1, my);
      atomicMaxF(mp + 2, mz);
      atomicMaxF(mp + 3, mw);
      if (lane == 0) atomicAdd(ws_cnt + curSeg, cnt);
    }
  };

  auto prefetch = [&](int t) {
    const int valid = min(TILE, rowsHere - t * TILE);
    const int vb = valid * D * 4;
    const char* g = (const char*)(x + (row0 + (long long)t * TILE) * D);
    char* l = (char*)&buf[t & 1][0];
#pragma unroll
    for (int j = 0; j < CHUNKS_PER_THREAD; ++j) {
      const int off = (tid + j * BLOCK) * 16;
      if (off < vb) async_copy_b128(g + off, l + off);
    }
  };

  prefetch(0);
  for (int t = 0; t < ntiles; ++t) {
    const bool haveNext = (t + 1 < ntiles);
    const bool nextFull = haveNext && (rowsHere - (t + 1) * TILE >= TILE);
    if (haveNext) prefetch(t + 1);
    // loads complete in order: waiting for <=CHUNKS outstanding guarantees tile t
    // landed while a full tile t+1 is still in flight. Partial next tile -> drain.
    if (nextFull)
      wait_async<CHUNKS_PER_THREAD>();
    else
      wait_async<0>();
    __syncthreads();

    const int valid = min(TILE, rowsHere - t * TILE);
    const float* bufc = &buf[t & 1][0];
    for (int rl = wave; rl < valid; rl += WAVES) {
      const long long row = row0 + (long long)t * TILE + rl;
      const int seg = batch[row];
      const float4 xv = *(const float4*)(bufc + rl * D + lane * 4);

      // 128-wide dot: per-lane 4-term partial + wave32 butterfly reduction
      float pd = fmaf(w4.x, xv.x, fmaf(w4.y, xv.y, fmaf(w4.z, xv.z, w4.w * xv.w)));
#pragma unroll
      for (int m = 16; m >= 1; m >>= 1) pd += __shfl_xor(pd, m, 32);
      const float gate = 1.0f / (1.0f + __expf(-(pd + bias)));

      if (seg != curSeg) {  // batch is sorted -> rare (once per ~977 rows)
        flush();
        curSeg = seg;
        sx = sy = sz = sw = 0.f;
        mx = my = mz = mw = -__builtin_inff();
        cnt = 0.f;
      }
      sx = fmaf(gate, xv.x, sx);
      sy = fmaf(gate, xv.y, sy);
      sz = fmaf(gate, xv.z, sz);
      sw = fmaf(gate, xv.w, sw);
      mx = fmaxf(mx, xv.x);
      my = fmaxf(my, xv.y);
      mz = fmaxf(mz, xv.z);
      mw = fmaxf(mw, xv.w);
      cnt += 1.f;
    }
    __syncthreads();
  }
  flush();
}

__global__ __launch_bounds__(2 * D) void afp_finalize_kernel(
    const float* __restrict__ ws_sum, const float* __restrict__ ws_max,
    const float* __restrict__ ws_cnt, float* __restrict__ out) {
  const int g = blockIdx.x;
  const int c = threadIdx.x;  // 0..255
  float v;
  if (c < D) {
    v = ws_max[g * D + c];
  } else {
    const float cn = fmaxf(ws_cnt[g], 1.0f);
    v = ws_sum[g * D + (c - D)] / cn;
  }
  out[g * (2 * D) + c] = v;
}

extern "C" void kernel_launch(void* const* d_in, const int* in_sizes, int n_in,
                              void* d_out, int out_size, void* d_ws, size_t ws_size,
                              hipStream_t stream) {
  const float* x = (const float*)d_in[0];
  // d_in[1] = edge_index: unused by the reference computation
  const int* batch = (const int*)d_in[2];
  const float* attn_w = (const float*)d_in[3];
  const float* attn_b = (const float*)d_in[4];
  float* out = (float*)d_out;

  const int N = in_sizes[0] / D;

  float* ws_sum = (float*)d_ws;                 // [G*D]
  float* ws_max = ws_sum + NUM_GRAPHS * D;      // [G*D]
  float* ws_cnt = ws_max + NUM_GRAPHS * D;      // [G]

  const int initElems = 2 * NUM_GRAPHS * D + NUM_GRAPHS;
  afp_init_kernel<<<(initElems + BLOCK - 1) / BLOCK, BLOCK, 0, stream>>>(ws_sum,
                                                                         initElems);

  const int nblk = (N + ROWS_PER_BLOCK - 1) / ROWS_PER_BLOCK;
  afp_pool_kernel<<<nblk, BLOCK, 0, stream>>>(x, batch, attn_w, attn_b, ws_sum,
                                              ws_max, ws_cnt, N);

  afp_finalize_kernel<<<NUM_GRAPHS, 2 * D, 0, stream>>>(ws_sum, ws_max, ws_cnt, out);
}